// LinearChainCRF_34626026340955
// MI455X (gfx1250) — compile-verified
//
#include <hip/hip_runtime.h>
#include <stdint.h>

#define Bx 256
#define Sx 512
#define Tx 128
#define PAD 132   // LDS row stride (floats); 132 mod 64 = 4 -> conflict-free b64 loads

typedef __attribute__((ext_vector_type(2))) float v2f;
typedef __attribute__((ext_vector_type(8))) float v8f;

// ---------------------------------------------------------------------------
// Kernel 0: E[i][j] = exp(transitions[i][j])  (step-invariant, computed once)
// ---------------------------------------------------------------------------
__global__ void crf_exp_kernel(const float* __restrict__ trans,
                               float* __restrict__ Eexp) {
    int i = blockIdx.x * blockDim.x + threadIdx.x;
    if (i < Tx * Tx) Eexp[i] = __expf(trans[i]);
}

// ---------------------------------------------------------------------------
// Kernel 1: numerator (gold-path score). One wave32 per batch element.
// ---------------------------------------------------------------------------
__global__ void crf_score_kernel(const float* __restrict__ emis,
                                 const int* __restrict__ tags,
                                 const uint8_t* __restrict__ mask,
                                 const float* __restrict__ startT,
                                 const float* __restrict__ endT,
                                 const float* __restrict__ trans,
                                 float* __restrict__ num) {
    const int lane = threadIdx.x & 31;
    const int b    = blockIdx.x * (blockDim.x >> 5) + (threadIdx.x >> 5);
    const size_t tb = (size_t)b * Sx;

    float part = 0.f;
    int   msum = 0;
    for (int t = lane; t < Sx; t += 32) {
        const uint8_t mk = mask[tb + t];
        msum += mk ? 1 : 0;
        if (t >= 1) {
            const uint8_t mkp = mask[tb + t - 1];
            const int tp = mkp ? tags[tb + t - 1] : 0;
            const int tc = mk  ? tags[tb + t]     : 0;
            float term = trans[tp * Tx + tc] + emis[(tb + t) * (size_t)Tx + tc];
            part += mk ? term : 0.f;
        }
    }
#pragma unroll
    for (int off = 1; off < 32; off <<= 1) {
        part += __shfl_xor(part, off, 32);
        msum += __shfl_xor(msum, off, 32);
    }
    if (lane == 0) {
        const int tg0 = mask[tb] ? tags[tb] : 0;
        float sc = startT[tg0] + emis[tb * (size_t)Tx + tg0] + part;
        const int len  = msum - 1;
        const int last = mask[tb + len] ? tags[tb + len] : 0;
        num[b] = sc + endT[last];
    }
}

// ---------------------------------------------------------------------------
// Kernel 2: forward algorithm (normalizer) + final subtraction.
// One block = one 16-row batch tile; 8 waves; fp32 WMMA 16x16x4 per step.
// Emissions are software-pipelined one step ahead (loads for t+1 issue while
// the WMMA chain for step t executes).
// ---------------------------------------------------------------------------
__global__ __launch_bounds__(256)
void crf_forward_kernel(const float* __restrict__ emis,
                        const uint8_t* __restrict__ mask,
                        const float* __restrict__ startT,
                        const float* __restrict__ endT,
                        const float* __restrict__ Eexp,
                        const float* __restrict__ num,
                        float* __restrict__ out) {
    __shared__ float   alpha_lds[16 * PAD];
    __shared__ float   ea_lds[16 * PAD];
    __shared__ float   mrow_lds[16];
    __shared__ uint8_t mk_lds[16];

    const int tid  = threadIdx.x;
    const int b0   = blockIdx.x * 16;
    const int lane = tid & 31;
    const int wave = tid >> 5;

    // phase-A mapping: 16 threads per row, 8 columns each
    const int r  = tid >> 4;          // row 0..15
    const int c0 = (tid & 15) * 8;    // column start

    // phase-B/C mapping: wave owns a 16-column tile of the output
    const int col   = wave * 16 + (lane & 15);
    const int koff  = (lane >> 4) * 2;               // ISA fp32 A/B striping
    const int abase = (lane & 15) * PAD + koff;      // A-operand LDS base
    const int mrow0 = (lane >> 4) * 8;               // C/D row base for this lane

    // Preload this wave's full 128x16 slice of E = exp(trans) into 64 VGPRs.
    // B 4x16 fp32 layout: VGPR0 = {K=0 | K=2}, VGPR1 = {K=1 | K=3} across lane halves.
    v2f breg[32];
#pragma unroll
    for (int kk = 0; kk < 32; ++kk) {
        const int k0 = kk * 4 + koff;
        breg[kk].x = Eexp[k0 * Tx + col];
        breg[kk].y = Eexp[(k0 + 1) * Tx + col];
    }

    // alpha0 = start_transitions + emissions[:, 0, :]
#pragma unroll
    for (int i = 0; i < 8; ++i) {
        const int c = c0 + i;
        alpha_lds[r * PAD + c] = startT[c] + emis[((size_t)(b0 + r) * Sx) * Tx + c];
    }

    // prime the emission pipeline with step t=1
    float em_cur[8], em_nxt[8];
#pragma unroll
    for (int v = 0; v < 8; ++v) {
        const int M = mrow0 + v;
        em_cur[v] = emis[((size_t)(b0 + M) * Sx + 1) * Tx + col];
    }
    __syncthreads();

    for (int t = 1; t < Sx; ++t) {
        // ---- phase A: row max + exp(alpha - max) into A-operand layout ----
        float av[8];
#pragma unroll
        for (int i = 0; i < 8; ++i) av[i] = alpha_lds[r * PAD + c0 + i];
        float mx = av[0];
#pragma unroll
        for (int i = 1; i < 8; ++i) mx = fmaxf(mx, av[i]);
#pragma unroll
        for (int off = 1; off < 16; off <<= 1) mx = fmaxf(mx, __shfl_xor(mx, off, 32));
#pragma unroll
        for (int i = 0; i < 8; ++i) ea_lds[r * PAD + c0 + i] = __expf(av[i] - mx);
        if ((tid & 15) == 0) {
            mrow_lds[r] = mx;
            mk_lds[r]   = mask[(size_t)(b0 + r) * Sx + t];
        }
        __syncthreads();

        // prefetch emissions for step t+1; latency hidden under the WMMA chain
        if (t + 1 < Sx) {
#pragma unroll
            for (int v = 0; v < 8; ++v) {
                const int M = mrow0 + v;
                em_nxt[v] = emis[((size_t)(b0 + M) * Sx + (t + 1)) * Tx + col];
            }
        }

        // ---- phase B: S(16x16 tile) = EA(16x128) @ E(128x16) via WMMA fp32 ----
        v8f acc = {};
#pragma unroll
        for (int kk = 0; kk < 32; ++kk) {
            v2f a = *(const v2f*)&ea_lds[abase + kk * 4];   // ds_load_2addr_b64
            acc = __builtin_amdgcn_wmma_f32_16x16x4_f32(
                false, a, false, breg[kk], (short)0, acc, false, false);
        }

        // ---- phase C: alpha' = rowmax + log(S) + em  (mask-gated) ----
#pragma unroll
        for (int v = 0; v < 8; ++v) {
            const int M  = mrow0 + v;
            const float nv  = mrow_lds[M] + __logf(acc[v]) + em_cur[v];
            const float old = alpha_lds[M * PAD + col];
            alpha_lds[M * PAD + col] = mk_lds[M] ? nv : old;
        }
        __syncthreads();

#pragma unroll
        for (int v = 0; v < 8; ++v) em_cur[v] = em_nxt[v];
    }

    // ---- epilogue: Z = logsumexp(alpha + end_transitions); out = num - Z ----
    float vv[8];
#pragma unroll
    for (int i = 0; i < 8; ++i) vv[i] = alpha_lds[r * PAD + c0 + i] + endT[c0 + i];
    float mx = vv[0];
#pragma unroll
    for (int i = 1; i < 8; ++i) mx = fmaxf(mx, vv[i]);
#pragma unroll
    for (int off = 1; off < 16; off <<= 1) mx = fmaxf(mx, __shfl_xor(mx, off, 32));
    float s = 0.f;
#pragma unroll
    for (int i = 0; i < 8; ++i) s += __expf(vv[i] - mx);
#pragma unroll
    for (int off = 1; off < 16; off <<= 1) s += __shfl_xor(s, off, 32);
    if ((tid & 15) == 0) out[b0 + r] = num[b0 + r] - (mx + __logf(s));
}

// ---------------------------------------------------------------------------
extern "C" void kernel_launch(void* const* d_in, const int* in_sizes, int n_in,
                              void* d_out, int out_size, void* d_ws, size_t ws_size,
                              hipStream_t stream) {
    const float*   emis   = (const float*)d_in[0];
    const int*     tags   = (const int*)d_in[1];
    const uint8_t* mask   = (const uint8_t*)d_in[2];
    const float*   startT = (const float*)d_in[3];
    const float*   endT   = (const float*)d_in[4];
    const float*   trans  = (const float*)d_in[5];
    float*         out    = (float*)d_out;

    float* Eexp = (float*)d_ws;          // 128*128 floats
    float* num  = Eexp + Tx * Tx;        // 256 floats

    crf_exp_kernel<<<(Tx * Tx + 255) / 256, 256, 0, stream>>>(trans, Eexp);
    crf_score_kernel<<<Bx / 8, 256, 0, stream>>>(emis, tags, mask, startT, endT, trans, num);
    crf_forward_kernel<<<Bx / 16, 256, 0, stream>>>(emis, mask, startT, endT, Eexp, num, out);
}